// SA_39152921870803
// MI455X (gfx1250) — compile-verified
//
#include <hip/hip_runtime.h>
#include <math.h>

// ---------------------------------------------------------------------------
// Slot-attention forward for MI455X (gfx1250), fp32 end-to-end using
// V_WMMA_F32_16X16X4_F32 for the K/V, dots and updates GEMMs.
//   Kernel 1: LN(x) + K/V projection (WMMA), one pass over x (HBM-bound).
//   Kernel 2: per-batch block (8 waves): 9 slot-attention iterations
//             (dots + softmax + updates via WMMA; v tiles staged into LDS
//             with GLOBAL_LOAD_ASYNC_TO_LDS_B128 overlapping the dots WMMAs),
//             GRU + MLP, agglomerative clustering, outputs.
// k/v (33.5 MB each) stay resident in the 192 MB L2 across all 9 iterations.
// ---------------------------------------------------------------------------

typedef __attribute__((ext_vector_type(2))) float v2f;
typedef __attribute__((ext_vector_type(8))) float v8f;
typedef int v4i __attribute__((vector_size(4 * sizeof(int))));
typedef __attribute__((address_space(1))) v4i* gpu_v4i_ptr;   // global
typedef __attribute__((address_space(3))) v4i* lds_v4i_ptr;   // LDS

#define B_    32
#define N_    4096
#define DIN_  256
#define DS_   64
#define NS_   8
#define NITER_ 9           // 8 fixed-point iters + 1 final (mask is all-true)
#define EPS_  1e-8f
#define THRESH_ 0.5f
#define SCALE_ 0.0625f     // DIN^-0.5 = 1/16

#define SLOTS_SZ (B_ * NS_ * DS_)        // 16384
#define ATTN_SZ  ((long)B_ * N_ * NS_)   // 1048576

#if defined(__has_builtin)
#if __has_builtin(__builtin_amdgcn_global_load_async_to_lds_b128)
#define ASYNC_LDS 1
#endif
#if __has_builtin(__builtin_amdgcn_s_wait_asynccnt)
#define HAVE_WAIT_ASYNC 1
#endif
#endif

static __device__ __forceinline__ v8f wmma_f32(v2f a, v2f b, v8f c) {
  // D(16x16,f32) = A(16x4,f32) * B(4x16,f32) + C
  return __builtin_amdgcn_wmma_f32_16x16x4_f32(false, a, false, b,
                                               (short)0, c, false, false);
}

// 16B global -> LDS copy; async (ASYNCcnt) when the toolchain exposes it.
static __device__ __forceinline__ void g2l_16B(const float* g, float* l) {
#ifdef ASYNC_LDS
  __builtin_amdgcn_global_load_async_to_lds_b128(
      (gpu_v4i_ptr)(void*)(const_cast<float*>(g)),
      (lds_v4i_ptr)(void*)l, 0, 0);
#else
  *(float4*)l = *(const float4*)g;
#endif
}
static __device__ __forceinline__ void g2l_wait() {
#if defined(ASYNC_LDS) && defined(HAVE_WAIT_ASYNC)
  __builtin_amdgcn_s_wait_asynccnt(0);
#endif
}

// ---------------------------------------------------------------------------
// Kernel 1: xn = LN(x); k = xn @ Wk^T; v = xn @ Wv^T
// grid = B*N/16 blocks, 256 threads (8 waves). Wave w<4 -> K tile w,
// wave w>=4 -> V tile w-4. Each wave does 64 WMMA k-steps over K=256.
// ---------------------------------------------------------------------------
__global__ __launch_bounds__(256) void ln_kv_kernel(
    const float* __restrict__ x,  const float* __restrict__ Wk,
    const float* __restrict__ Wv, const float* __restrict__ lnw,
    const float* __restrict__ lnb, float* __restrict__ kOut,
    float* __restrict__ vOut) {
  __shared__ float xn[16][260];   // stride 260 -> rows land on distinct banks
  __shared__ float red[256];
  __shared__ float red2[256];
  __shared__ float sMean[16];
  __shared__ float sRstd[16];

  const int tid = threadIdx.x;
  const int row = tid >> 4;            // 0..15
  const int seg = (tid & 15) * 16;     // 16 contiguous elems per thread
  const long g = (long)blockIdx.x * 16 + row;
  const float* xr = x + g * DIN_ + seg;

  float xv[16];
  float s = 0.f, sq = 0.f;
#pragma unroll
  for (int e = 0; e < 16; ++e) { float t = xr[e]; xv[e] = t; s += t; sq += t * t; }
  red[tid] = s; red2[tid] = sq;
  __syncthreads();
  if (tid < 16) {
    float a = 0.f, bq = 0.f;
    for (int t = 0; t < 16; ++t) { a += red[tid * 16 + t]; bq += red2[tid * 16 + t]; }
    float m = a * (1.0f / 256.0f);
    float v = bq * (1.0f / 256.0f) - m * m;
    sMean[tid] = m;
    sRstd[tid] = rsqrtf(v + 1e-5f);
  }
  __syncthreads();
  {
    float m = sMean[row], r = sRstd[row];
#pragma unroll
    for (int e = 0; e < 16; ++e)
      xn[row][seg + e] = (xv[e] - m) * r * lnw[seg + e] + lnb[seg + e];
  }
  __syncthreads();

  const int wid  = tid >> 5;           // 0..7
  const int lane = tid & 31;
  const float* W = (wid >= 4) ? Wv : Wk;
  float*       O = (wid >= 4) ? vOut : kOut;
  const int n0   = (wid & 3) * 16;
  const int l15  = lane & 15;
  const int kp   = (lane >> 4) * 2;    // 0 or 2

  v8f acc = {};
  for (int k0 = 0; k0 < DIN_; k0 += 4) {
    v2f a, b;
    // A 16x4: lane l15 = row M, VGPR pair holds K = kp, kp+1
    a.x = xn[l15][k0 + kp];
    a.y = xn[l15][k0 + kp + 1];
    // B 4x16: B[c][n] = W[n0+n][c]
    const float* wp = W + (n0 + l15) * DIN_ + k0 + kp;
    b.x = wp[0]; b.y = wp[1];
    acc = wmma_f32(a, b, acc);
  }
  // D 16x16 f32: VGPR r, lane L -> M = r + (L>>4)*8, Ncol = L&15
  const long base = (long)blockIdx.x * 16;
  const int  col  = n0 + l15;
  const int  moff = (lane >> 4) * 8;
#pragma unroll
  for (int r = 0; r < 8; ++r)
    O[(base + moff + r) * DS_ + col] = acc[r];
}

// ---------------------------------------------------------------------------
// Kernel 2: per-batch slot attention (9 iterations) + clustering + outputs.
// grid = B blocks, 256 threads (8 waves). Each wave owns j-tiles strided 128.
// ---------------------------------------------------------------------------
__global__ __launch_bounds__(256) void slot_kernel(
    const float* __restrict__ kG,   const float* __restrict__ vG,
    const float* __restrict__ semb, const float* __restrict__ Wq,
    const float* __restrict__ gWih, const float* __restrict__ gWhh,
    const float* __restrict__ gbih, const float* __restrict__ gbhh,
    const float* __restrict__ W1,   const float* __restrict__ b1,
    const float* __restrict__ W2,   const float* __restrict__ b2,
    const float* __restrict__ lsw,  const float* __restrict__ lsb,
    const float* __restrict__ lpw,  const float* __restrict__ lpb,
    float* __restrict__ attnG, float* __restrict__ out) {
  __shared__ float slots[8][64];
  __shared__ float rowtmp[8][64];     // LN(slots) then later LN(h) (pre)
  __shared__ float qls[16][68];       // rows 8..15 stay zero (A-matrix pad)
  __shared__ float upd[8][64];
  __shared__ float hnl[8][64];
  __shared__ float wld[8][16][17];    // per-wave softmax tile, rows 8..15 zero
  __shared__ float vt[8][16][68];     // per-wave v tile (B128-aligned rows);
                                      // aliased as h1[8][256] in MLP phase
  __shared__ float vsum[64];
  __shared__ float vpart[4][64];
  __shared__ float rsT[8];
  __shared__ float adjA[8][8];
  __shared__ float adjB[8][8];
  __shared__ float cm[8][8];
  __shared__ float cw[8][8];
  __shared__ float nrmv[8];
  __shared__ int   cnts;

  float* h1f = &vt[0][0][0];          // [8][256] alias (phase-disjoint)

  const int tid  = threadIdx.x;
  const int wid  = tid >> 5;          // 0..7
  const int lane = tid & 31;
  const int l15  = lane & 15;
  const int kp   = (lane >> 4) * 2;
  const int b    = blockIdx.x;

  const float* kB = kG + (long)b * N_ * DS_;
  const float* vB = vG + (long)b * N_ * DS_;
  float* attnB = attnG + (long)b * NS_ * N_;

  for (int i = tid; i < 512; i += 256) slots[i >> 6][i & 63] = semb[i];
  for (int i = tid; i < 16 * 68; i += 256) (&qls[0][0])[i] = 0.0f;
  for (int i = tid; i < 8 * 16 * 17; i += 256) (&wld[0][0][0])[i] = 0.0f;
  {  // vsum[c] = sum_j v[j][c]
    int c = tid & 63, part = tid >> 6;      // 4 partials per column
    float s = 0.f;
    for (int j = part * 1024; j < part * 1024 + 1024; ++j) s += vB[j * DS_ + c];
    vpart[part][c] = s;
  }
  __syncthreads();
  if (tid < 64) vsum[tid] = vpart[0][tid] + vpart[1][tid] + vpart[2][tid] + vpart[3][tid];
  __syncthreads();

  for (int it = 0; it < NITER_; ++it) {
    // ---- q = LN(slots) @ Wq^T ------------------------------------------
    if (tid < 8) {
      float m = 0.f;
      for (int c = 0; c < 64; ++c) m += slots[tid][c];
      m *= (1.0f / 64.0f);
      float v = 0.f;
      for (int c = 0; c < 64; ++c) { float d0 = slots[tid][c] - m; v += d0 * d0; }
      float r = rsqrtf(v * (1.0f / 64.0f) + 1e-5f);
      for (int c = 0; c < 64; ++c) rowtmp[tid][c] = (slots[tid][c] - m) * r * lsw[c] + lsb[c];
    }
    __syncthreads();
    for (int idx = tid; idx < 512; idx += 256) {
      int i = idx >> 6, d0 = idx & 63;
      float s = 0.f;
      const float* wr = Wq + d0 * 64;
      for (int c = 0; c < 64; ++c) s += rowtmp[i][c] * wr[c];
      qls[i][d0] = s;
    }
    for (int idx = tid; idx < 512; idx += 256) upd[idx >> 6][idx & 63] = 0.0f;
    if (tid < 8) rsT[tid] = 0.0f;
    __syncthreads();

    // ---- single pass over N: dots -> softmax(slots) -> updates ---------
    const bool store_attn = (it == NITER_ - 1);
    float rs[8];
#pragma unroll
    for (int r = 0; r < 8; ++r) rs[r] = 0.0f;
    v8f u0 = {}, u1 = {}, u2 = {}, u3 = {};

    for (int j0 = wid * 16; j0 < N_; j0 += 128) {
      // stage this wave's 16x64 v tile into LDS (async, overlaps dots WMMAs)
#pragma unroll
      for (int t = 0; t < 8; ++t) {
        int chunk = t * 32 + lane;        // 256 chunks of 16B
        int row = chunk >> 4, c4 = (chunk & 15) * 4;
        g2l_16B(vB + (long)(j0 + row) * DS_ + c4, &vt[wid][row][c4]);
      }
      // dots tile: D = q(16x64, rows 8..15 zero) @ k^T(64x16)
      v8f d = {};
      for (int c0 = 0; c0 < 64; c0 += 4) {
        v2f a, bb;
        a.x = qls[l15][c0 + kp];
        a.y = qls[l15][c0 + kp + 1];
        const float* kr = kB + (long)(j0 + l15) * DS_ + c0 + kp;
        bb.x = kr[0]; bb.y = kr[1];
        d = wmma_f32(a, bb, d);
      }
      // per-lane softmax over the 8 slot rows (lane l<16 owns column j0+l)
      float dv[8];
#pragma unroll
      for (int r = 0; r < 8; ++r) dv[r] = d[r] * SCALE_;
      float mx = dv[0];
#pragma unroll
      for (int r = 1; r < 8; ++r) mx = fmaxf(mx, dv[r]);
      float ssum = 0.f;
#pragma unroll
      for (int r = 0; r < 8; ++r) { dv[r] = expf(dv[r] - mx); ssum += dv[r]; }
      float inv = 1.0f / ssum;
#pragma unroll
      for (int r = 0; r < 8; ++r) { dv[r] *= inv; rs[r] += dv[r]; }

      if (lane < 16) {
#pragma unroll
        for (int r = 0; r < 8; ++r) wld[wid][r][lane] = dv[r];
        if (store_attn) {
#pragma unroll
          for (int r = 0; r < 8; ++r) attnB[r * N_ + j0 + lane] = dv[r];  // attn_vis
        }
      }
      g2l_wait();   // v tile resident in LDS before updates WMMAs
      // updates tile: U += W(16xN tile) @ v(N tile x 64), K-steps of 4
#pragma unroll
      for (int kk = 0; kk < 4; ++kk) {
        v2f a;
        a.x = wld[wid][l15][kk * 4 + kp];
        a.y = wld[wid][l15][kk * 4 + kp + 1];
        const float* r0 = &vt[wid][kk * 4 + kp][0];
        const float* r1 = &vt[wid][kk * 4 + kp + 1][0];
        v2f bb;
        bb.x = r0[l15];      bb.y = r1[l15];      u0 = wmma_f32(a, bb, u0);
        bb.x = r0[16 + l15]; bb.y = r1[16 + l15]; u1 = wmma_f32(a, bb, u1);
        bb.x = r0[32 + l15]; bb.y = r1[32 + l15]; u2 = wmma_f32(a, bb, u2);
        bb.x = r0[48 + l15]; bb.y = r1[48 + l15]; u3 = wmma_f32(a, bb, u3);
      }
    }

    // reduce row sums across lanes 0..15 (xor masks stay inside half-wave)
#pragma unroll
    for (int r = 0; r < 8; ++r) {
      float t = rs[r];
      t += __shfl_xor(t, 1); t += __shfl_xor(t, 2);
      t += __shfl_xor(t, 4); t += __shfl_xor(t, 8);
      if (lane == 0) atomicAdd(&rsT[r], t);
    }
    if (lane < 16) {
#pragma unroll
      for (int r = 0; r < 8; ++r) {
        atomicAdd(&upd[r][lane],      u0[r]);
        atomicAdd(&upd[r][16 + lane], u1[r]);
        atomicAdd(&upd[r][32 + lane], u2[r]);
        atomicAdd(&upd[r][48 + lane], u3[r]);
      }
    }
    __syncthreads();
    // updates_i = (U_i + EPS*vsum) / (rowsum_i + N*EPS)
    for (int idx = tid; idx < 512; idx += 256) {
      int i = idx >> 6, c = idx & 63;
      upd[i][c] = (upd[i][c] + EPS_ * vsum[c]) / (rsT[i] + (float)N_ * EPS_);
    }
    __syncthreads();

    // ---- GRU cell ------------------------------------------------------
    for (int idx = tid; idx < 512; idx += 256) {
      int i = idx >> 6, c = idx & 63;
      float ir = gbih[c], iz = gbih[64 + c], inn = gbih[128 + c];
      float hr = gbhh[c], hz = gbhh[64 + c], hnn = gbhh[128 + c];
      const float* wri = gWih + c * 64;
      const float* wzi = gWih + (64 + c) * 64;
      const float* wni = gWih + (128 + c) * 64;
      const float* wrh = gWhh + c * 64;
      const float* wzh = gWhh + (64 + c) * 64;
      const float* wnh = gWhh + (128 + c) * 64;
      for (int cc = 0; cc < 64; ++cc) {
        float uv = upd[i][cc], hv = slots[i][cc];
        ir += uv * wri[cc]; iz += uv * wzi[cc]; inn += uv * wni[cc];
        hr += hv * wrh[cc]; hz += hv * wzh[cc]; hnn += hv * wnh[cc];
      }
      float r = 1.0f / (1.0f + expf(-(ir + hr)));
      float z = 1.0f / (1.0f + expf(-(iz + hz)));
      float n = tanhf(inn + r * hnn);
      hnl[i][c] = (1.0f - z) * n + z * slots[i][c];
    }
    __syncthreads();

    // ---- residual MLP: slots = h + W2 @ relu(W1 @ LN(h)) ---------------
    if (tid < 8) {
      float m = 0.f;
      for (int c = 0; c < 64; ++c) m += hnl[tid][c];
      m *= (1.0f / 64.0f);
      float v = 0.f;
      for (int c = 0; c < 64; ++c) { float d0 = hnl[tid][c] - m; v += d0 * d0; }
      float r = rsqrtf(v * (1.0f / 64.0f) + 1e-5f);
      for (int c = 0; c < 64; ++c) rowtmp[tid][c] = (hnl[tid][c] - m) * r * lpw[c] + lpb[c];
    }
    __syncthreads();
    for (int idx = tid; idx < 2048; idx += 256) {   // h1 aliases vt (safe: barrier)
      int i = idx >> 8, m = idx & 255;
      float s = b1[m];
      const float* w = W1 + m * 64;
      for (int c = 0; c < 64; ++c) s += rowtmp[i][c] * w[c];
      h1f[i * 256 + m] = fmaxf(s, 0.0f);
    }
    __syncthreads();
    for (int idx = tid; idx < 512; idx += 256) {
      int i = idx >> 6, c = idx & 63;
      float s = b2[c];
      const float* w = W2 + c * 256;
      for (int m = 0; m < 256; ++m) s += h1f[i * 256 + m] * w[m];
      slots[i][c] = hnl[i][c] + s;
    }
    __syncthreads();
  }  // iterations

  // ---- agglomerative clustering (single linkage, 8x8) ------------------
  if (tid < 8) {
    float s = 0.f;
    for (int c = 0; c < 64; ++c) { float t = slots[tid][c]; s += t * t; }
    nrmv[tid] = rsqrtf(fmaxf(s, 1e-30f));
  }
  __syncthreads();
  if (tid < 64) {
    int i = tid >> 3, j = tid & 7;
    float s = 0.f;
    for (int c = 0; c < 64; ++c) s += slots[i][c] * slots[j][c];
    float dist = 1.0f - s * nrmv[i] * nrmv[j];
    adjA[i][j] = (dist < THRESH_) ? 1.0f : 0.0f;
  }
  __syncthreads();
  for (int t = 0; t < 3; ++t) {  // transitive closure: ceil(log2(8)) squarings
    if (tid < 64) {
      int i = tid >> 3, j = tid & 7;
      float s = 0.f;
      for (int m = 0; m < 8; ++m) s += adjA[i][m] * adjA[m][j];
      adjB[i][j] = fminf(s, 1.0f);
    }
    __syncthreads();
    if (tid < 64) adjA[tid >> 3][tid & 7] = adjB[tid >> 3][tid & 7];
    __syncthreads();
  }
  if (tid < 8) {  // cumsum over rows per column tid; keep first representative
    float cs = 0.f;
    for (int n = 0; n < 8; ++n) {
      cs += adjA[n][tid];
      cm[n][tid] = (cs <= 1.0f) ? adjA[n][tid] : 0.0f;
    }
  }
  __syncthreads();
  if (tid == 0) {
    int cnt = 0;
    for (int n = 0; n < 8; ++n) {
      float s = 0.f;
      for (int m = 0; m < 8; ++m) s += cm[n][m];
      if (s > 0.f) cnt++;
    }
    cnts = cnt;
  }
  __syncthreads();
  {
    bool useEye = (cnts < 2);
    if (tid < 64 && useEye)
      cm[tid >> 3][tid & 7] = ((tid >> 3) == (tid & 7)) ? 1.0f : 0.0f;
  }
  __syncthreads();
  if (tid < 8) {
    float s = 0.f;
    for (int m = 0; m < 8; ++m) s += cm[tid][m];
    float inv = 1.0f / fmaxf(s, 1.0f);
    for (int m = 0; m < 8; ++m) cw[tid][m] = cm[tid][m] * inv;
    out[SLOTS_SZ + ATTN_SZ + b * 8 + tid] = (s > 0.f) ? 1.0f : 0.0f;  // slot_mask
  }
  __syncthreads();
  for (int idx = tid; idx < 512; idx += 256) {  // means = cweight @ slots
    int n = idx >> 6, dcol = idx & 63;
    float s = 0.f;
    for (int m = 0; m < 8; ++m) s += cw[n][m] * slots[m][dcol];
    out[(long)b * 512 + idx] = s;
  }
  for (int j = tid; j < N_; j += 256) {  // attn_map[b,j,n] = sum_m cm[n][m]*attn[m][j]
    float av[8];
#pragma unroll
    for (int m = 0; m < 8; ++m) av[m] = attnB[m * N_ + j];
    float* op = out + SLOTS_SZ + ((long)b * N_ + j) * 8;
#pragma unroll
    for (int n = 0; n < 8; ++n) {
      float s = 0.f;
#pragma unroll
      for (int m = 0; m < 8; ++m) s += cm[n][m] * av[m];
      op[n] = s;
    }
  }
}

// ---------------------------------------------------------------------------
extern "C" void kernel_launch(void* const* d_in, const int* in_sizes, int n_in,
                              void* d_out, int out_size, void* d_ws, size_t ws_size,
                              hipStream_t stream) {
  (void)in_sizes; (void)n_in; (void)out_size; (void)ws_size;
  const float* x    = (const float*)d_in[0];
  const float* semb = (const float*)d_in[1];
  const float* Wk   = (const float*)d_in[2];
  const float* Wv   = (const float*)d_in[3];
  const float* Wq   = (const float*)d_in[4];
  const float* gWih = (const float*)d_in[5];
  const float* gWhh = (const float*)d_in[6];
  const float* gbih = (const float*)d_in[7];
  const float* gbhh = (const float*)d_in[8];
  const float* W1   = (const float*)d_in[9];
  const float* b1   = (const float*)d_in[10];
  const float* W2   = (const float*)d_in[11];
  const float* b2   = (const float*)d_in[12];
  const float* lnw  = (const float*)d_in[13];
  const float* lnb  = (const float*)d_in[14];
  const float* lsw  = (const float*)d_in[15];
  const float* lsb  = (const float*)d_in[16];
  const float* lpw  = (const float*)d_in[17];
  const float* lpb  = (const float*)d_in[18];

  float* ws = (float*)d_ws;
  float* kP = ws;                                  // [B, N, 64]
  float* vP = kP + (size_t)B_ * N_ * DS_;          // [B, N, 64]
  float* aP = vP + (size_t)B_ * N_ * DS_;          // [B, 8, N] final attn_vis
  float* outF = (float*)d_out;

  ln_kv_kernel<<<dim3(B_ * N_ / 16), dim3(256), 0, stream>>>(
      x, Wk, Wv, lnw, lnb, kP, vP);
  slot_kernel<<<dim3(B_), dim3(256), 0, stream>>>(
      kP, vP, semb, Wq, gWih, gWhh, gbih, gbhh, W1, b1, W2, b2,
      lsw, lsb, lpw, lpb, aP, outF);
}